// SCN_without_pool_60842506715652
// MI455X (gfx1250) — compile-verified
//
#include <hip/hip_runtime.h>
#include <math.h>

// ---------------------------------------------------------------------------
// SCN_without_pool for MI455X (gfx1250, wave32).
// GEMMs (fm @ w_l + b) run on V_WMMA_F32_16X16X4_F32 (exact fp32 — workload is
// latency-bound: ~3.75 GFLOP, ~100MB traffic, fits in 192MB L2).
// GEMM uses 2x4 register blocking per wave (32x64 output, 8 accumulators) for
// 2.7x operand-load reuse, plus global_prefetch on the streamed weight panel.
// ---------------------------------------------------------------------------

#define EPS_    1e-12f
#define BN_EPS_ 1e-5f
#define NV      1024   // vertices per batch
#define NB      2      // batch
#define KNN     16     // neighbors
#define FLT_BIG 3.402823466e+38f

typedef float v2f __attribute__((ext_vector_type(2)));
typedef float v8f __attribute__((ext_vector_type(8)));

// ---------------- normalize_cols(dirs): (3, cols) -> (3, cols) -------------
__global__ void normalize_dirs_kernel(const float* __restrict__ w,
                                      float* __restrict__ o, int cols) {
    int t = blockIdx.x * blockDim.x + threadIdx.x;
    if (t >= cols) return;
    float a = w[t], b = w[cols + t], c = w[2 * cols + t];
    float inv = 1.f / fmaxf(sqrtf(a * a + b * b + c * c), EPS_);
    o[t] = a * inv; o[cols + t] = b * inv; o[2 * cols + t] = c * inv;
}

// ---------------- KNN + normalized neighbor directions ---------------------
// grid = 8 blocks x 256 thr; 4 blocks per batch; batch verts staged in LDS.
__global__ void knn_kernel(const float* __restrict__ verts,
                           int* __restrict__ idx, float* __restrict__ dn) {
    __shared__ float sv[NV * 3];
    int b = (int)blockIdx.x >> 2;
    const float* vb = verts + b * NV * 3;
    for (int i = threadIdx.x; i < NV * 3; i += blockDim.x) sv[i] = vb[i];
    __syncthreads();

    int v   = ((int)blockIdx.x & 3) * 256 + threadIdx.x;
    int gid = b * NV + v;
    float px = sv[v * 3], py = sv[v * 3 + 1], pz = sv[v * 3 + 2];

    float bd[KNN]; int bi[KNN];
#pragma unroll
    for (int i = 0; i < KNN; ++i) { bd[i] = FLT_BIG; bi[i] = 0; }

    for (int u = 0; u < NV; ++u) {
        if (u == v) continue;
        float dx = sv[u * 3] - px, dy = sv[u * 3 + 1] - py, dz = sv[u * 3 + 2] - pz;
        float d  = dx * dx + dy * dy + dz * dz;
        if (d < bd[KNN - 1]) {
            float cd = d; int ci = u;
#pragma unroll
            for (int j = 0; j < KNN; ++j) {          // register bubble-insert
                if (cd < bd[j]) {
                    float td = bd[j]; bd[j] = cd; cd = td;
                    int   ti = bi[j]; bi[j] = ci; ci = ti;
                }
            }
        }
    }
#pragma unroll
    for (int n = 0; n < KNN; ++n) {
        int u = bi[n];
        idx[gid * KNN + n] = u;
        float dx = sv[u * 3] - px, dy = sv[u * 3 + 1] - py, dz = sv[u * 3 + 2] - pz;
        float inv = 1.f / fmaxf(sqrtf(dx * dx + dy * dy + dz * dz), EPS_);
        float* p = dn + (gid * KNN + n) * 3;
        p[0] = dx * inv; p[1] = dy * inv; p[2] = dz * inv;
    }
}

// ---------------- conv_surface: -> fm0 (2048 x 32), relu -------------------
// one block (64 thr) per vertex; thread t = theta column k (k = s*32 + c)
__global__ void conv_surface_kernel(const float* __restrict__ dn,
                                    const float* __restrict__ dirs,   // (3,64) normalized
                                    float* __restrict__ fm) {
    __shared__ float dnl[KNN * 3];
    __shared__ float mx[64];
    int gid = blockIdx.x, t = threadIdx.x;
    if (t < KNN * 3) dnl[t] = dn[gid * KNN * 3 + t];
    __syncthreads();
    float d0 = dirs[t], d1 = dirs[64 + t], d2 = dirs[128 + t];
    float m = 0.f;  // relu(theta) >= 0
#pragma unroll
    for (int n = 0; n < KNN; ++n) {
        float th = fmaxf(dnl[n * 3] * d0 + dnl[n * 3 + 1] * d1 + dnl[n * 3 + 2] * d2, 0.f);
        m = fmaxf(m, th);
    }
    mx[t] = m;
    __syncthreads();
    if (t < 32) fm[gid * 32 + t] = fmaxf(mx[t] + mx[32 + t], 0.f);
}

// ---------------- WMMA f32 GEMM: out(M,N) = A(M,K) @ W(K,N) + bias ---------
// Register-blocked: each wave computes a 32x64 block = 2(M) x 4(N) 16x16
// tiles -> 8 v8f accumulators. Block = 4 waves stacked along M (128x64 tile).
// Requirements: K%4==0, N%64==0, M%128==0 (all layer shapes satisfy).
__global__ void wmma_gemm_bias(const float* __restrict__ A,
                               const float* __restrict__ W,
                               const float* __restrict__ bias,
                               float* __restrict__ out,
                               int M, int K, int N) {
    int wave = threadIdx.x >> 5;
    int lane = threadIdx.x & 31;
    int half = lane >> 4;                     // 0: lanes 0-15, 1: lanes 16-31
    int r    = lane & 15;

    int m0 = blockIdx.y * 128 + wave * 32;    // wave's 32 output rows
    int n0 = blockIdx.x * 64;                 // wave's 64 output cols

    const float* Ap0 = A + (size_t)(m0 + r) * K;        // rows of M-tile 0
    const float* Ap1 = A + (size_t)(m0 + 16 + r) * K;   // rows of M-tile 1

    v8f acc[2][4];
#pragma unroll
    for (int i = 0; i < 2; ++i)
#pragma unroll
        for (int j = 0; j < 4; ++j) acc[i][j] = (v8f){};

    for (int k = 0; k < K; k += 4) {
        int ka = k + 2 * half;                // this half-wave's K pair
        // A fragments (16x4 f32): v0 -> K=ka, v1 -> K=ka+1
        v2f a0, a1;
        a0.x = Ap0[ka]; a0.y = Ap0[ka + 1];
        a1.x = Ap1[ka]; a1.y = Ap1[ka + 1];
        // B fragments (4x16 f32), 4 N-tiles; lane holds column n0 + j*16 + r
        const float* Wp = W + (size_t)ka * N + n0 + r;
        // prefetch next k-block of the weight panel (emits global_prefetch_b8)
        __builtin_prefetch(Wp + 4 * N, 0, 1);
        v2f b0, b1, b2, b3;
        b0.x = Wp[0];  b0.y = Wp[N];
        b1.x = Wp[16]; b1.y = Wp[16 + N];
        b2.x = Wp[32]; b2.y = Wp[32 + N];
        b3.x = Wp[48]; b3.y = Wp[48 + N];

        acc[0][0] = __builtin_amdgcn_wmma_f32_16x16x4_f32(false, a0, false, b0, (short)0, acc[0][0], false, false);
        acc[0][1] = __builtin_amdgcn_wmma_f32_16x16x4_f32(false, a0, false, b1, (short)0, acc[0][1], false, false);
        acc[0][2] = __builtin_amdgcn_wmma_f32_16x16x4_f32(false, a0, false, b2, (short)0, acc[0][2], false, false);
        acc[0][3] = __builtin_amdgcn_wmma_f32_16x16x4_f32(false, a0, false, b3, (short)0, acc[0][3], false, false);
        acc[1][0] = __builtin_amdgcn_wmma_f32_16x16x4_f32(false, a1, false, b0, (short)0, acc[1][0], false, false);
        acc[1][1] = __builtin_amdgcn_wmma_f32_16x16x4_f32(false, a1, false, b1, (short)0, acc[1][1], false, false);
        acc[1][2] = __builtin_amdgcn_wmma_f32_16x16x4_f32(false, a1, false, b2, (short)0, acc[1][2], false, false);
        acc[1][3] = __builtin_amdgcn_wmma_f32_16x16x4_f32(false, a1, false, b3, (short)0, acc[1][3], false, false);
    }

    // C/D layout: vgpr q -> row (q + 8*half), lane r -> column
#pragma unroll
    for (int j = 0; j < 4; ++j) {
        int col = n0 + j * 16 + r;
        float bc = bias[col];
#pragma unroll
        for (int mt = 0; mt < 2; ++mt) {
#pragma unroll
            for (int q = 0; q < 8; ++q) {
                out[(size_t)(m0 + mt * 16 + half * 8 + q) * N + col] = acc[mt][j][q] + bc;
            }
        }
    }
}

// ---------------- conv layer aggregation -----------------------------------
// out[v,c] = center + max_n(relu(dn.d0c)*sup0) + max_n(relu(dn.d1c)*sup1)
__global__ void conv_aggregate_kernel(const float* __restrict__ fo,   // (2048, 3co)
                                      const float* __restrict__ dn,
                                      const int* __restrict__ idx,
                                      const float* __restrict__ dirs, // (3, 2co) normalized
                                      float* __restrict__ fm_out,     // (2048, co)
                                      int co, int relu_out) {
    __shared__ float dnl[KNN * 3];
    __shared__ int   nbr[KNN];
    int gid = blockIdx.x, t = threadIdx.x;
    int b = gid >> 10;
    if (t < KNN * 3) dnl[t] = dn[gid * KNN * 3 + t];
    if (t < KNN)     nbr[t] = idx[gid * KNN + t];
    __syncthreads();
    int n3 = 3 * co, k2 = 2 * co;
    for (int c = t; c < co; c += blockDim.x) {
        float d0x = dirs[c],          d0y = dirs[k2 + c],          d0z = dirs[2 * k2 + c];
        float d1x = dirs[co + c],     d1y = dirs[k2 + co + c],     d1z = dirs[2 * k2 + co + c];
        float m0 = -FLT_BIG, m1 = -FLT_BIG;
#pragma unroll 4
        for (int n = 0; n < KNN; ++n) {
            int row = (b << 10) + nbr[n];
            float x = dnl[n * 3], y = dnl[n * 3 + 1], z = dnl[n * 3 + 2];
            float th0 = fmaxf(x * d0x + y * d0y + z * d0z, 0.f);
            float th1 = fmaxf(x * d1x + y * d1y + z * d1z, 0.f);
            float s0 = fo[(size_t)row * n3 + co + c];
            float s1 = fo[(size_t)row * n3 + 2 * co + c];
            m0 = fmaxf(m0, th0 * s0);
            m1 = fmaxf(m1, th1 * s1);
        }
        float o = fo[(size_t)gid * n3 + c] + m0 + m1;
        if (relu_out) o = fmaxf(o, 0.f);
        fm_out[(size_t)gid * co + c] = o;
    }
}

// ---------------- global max pool over vertices ----------------------------
__global__ void maxpool_kernel(const float* __restrict__ fm, float* __restrict__ out) {
    int gid = blockIdx.x * blockDim.x + threadIdx.x;  // 2048 = (b,c)
    int b = gid >> 10, c = gid & 1023;
    const float* p = fm + (size_t)(b << 10) * 1024 + c;
    float m = -FLT_BIG;
    for (int v = 0; v < NV; ++v) m = fmaxf(m, p[v * 1024]);
    out[gid] = m;
}

// ---------------- MLP head: 1024 -> 256(BN,relu) -> out_dim ----------------
__global__ void head_kernel(const float* __restrict__ fg,
                            const float* __restrict__ w1, const float* __restrict__ b1,
                            const float* __restrict__ g,  const float* __restrict__ be,
                            const float* __restrict__ rm, const float* __restrict__ rv,
                            const float* __restrict__ w2, const float* __restrict__ b2,
                            float* __restrict__ out, int out_dim, int do_norm) {
    __shared__ float h[256];
    __shared__ float o[128];
    __shared__ float scale;
    int b = blockIdx.x, t = threadIdx.x;
    const float* x = fg + b * 1024;
    float acc = b1[t];
    for (int i = 0; i < 1024; ++i) acc += x[i] * w1[i * 256 + t];
    acc = (acc - rm[t]) * rsqrtf(rv[t] + BN_EPS_) * g[t] + be[t];
    h[t] = fmaxf(acc, 0.f);
    __syncthreads();
    if (t < out_dim) {
        float a = b2[t];
        for (int i = 0; i < 256; ++i) a += h[i] * w2[i * out_dim + t];
        o[t] = a;
    }
    __syncthreads();
    if (do_norm) {
        if (t == 0) {
            float s = 0.f;
            for (int i = 0; i < out_dim; ++i) s += o[i] * o[i];
            scale = 1.f / fmaxf(sqrtf(s), EPS_);
        }
        __syncthreads();
        if (t < out_dim) out[b * out_dim + t] = o[t] * scale;
    } else {
        if (t < out_dim) out[b * out_dim + t] = o[t];
    }
}

// ---------------------------------------------------------------------------
extern "C" void kernel_launch(void* const* d_in, const int* in_sizes, int n_in,
                              void* d_out, int out_size, void* d_ws, size_t ws_size,
                              hipStream_t stream) {
    (void)in_sizes; (void)n_in; (void)out_size; (void)ws_size;

    const float* verts = (const float*)d_in[0];
    const float* dirs_raw[5] = { (const float*)d_in[1],  (const float*)d_in[4],
                                 (const float*)d_in[7],  (const float*)d_in[10],
                                 (const float*)d_in[13] };
    const float* w[5]  = { nullptr, (const float*)d_in[2], (const float*)d_in[5],
                           (const float*)d_in[8], (const float*)d_in[11] };
    const float* bb[5] = { nullptr, (const float*)d_in[3], (const float*)d_in[6],
                           (const float*)d_in[9], (const float*)d_in[12] };

    // workspace layout (floats)
    float* ws = (float*)d_ws;
    float* dn  = ws;                           // 2*1024*16*3       =   98304
    int*   idx = (int*)(ws + 98304);           // 2*1024*16 ints    =   32768
    int dcols[5] = { 64, 128, 256, 512, 2048 };
    float* dirsn[5];
    {
        size_t off = 131072;
        for (int l = 0; l < 5; ++l) { dirsn[l] = ws + off; off += (size_t)3 * dcols[l]; }
    }
    float* fmA = ws + 140160;                  // 2048*1024 = 2097152
    float* fmB = fmA + 2097152;                // 2048*256  =  524288
    float* fo  = fmB + 524288;                 // 2048*3072 = 6291456  (~36.2 MB total)

    // 1. normalize direction kernels
    for (int l = 0; l < 5; ++l)
        normalize_dirs_kernel<<<(dcols[l] + 255) / 256, 256, 0, stream>>>(
            dirs_raw[l], dirsn[l], dcols[l]);

    // 2. KNN + neighbor dirs
    knn_kernel<<<8, 256, 0, stream>>>(verts, idx, dn);

    // 3. conv_surface -> fmA (2048 x 32), relu
    conv_surface_kernel<<<NB * NV, 64, 0, stream>>>(dn, dirsn[0], fmA);

    // 4. conv layers 1..4: WMMA GEMM then aggregate (ping-pong fmA/fmB)
    int ci[5] = { 0, 32, 64, 128, 256 };
    int co[5] = { 0, 64, 128, 256, 1024 };
    float* fin = fmA;
    float* fot = fmB;
    for (int l = 1; l <= 4; ++l) {
        int N = 3 * co[l];
        dim3 grid(N / 64, (NB * NV) / 128);    // 128x64 block tile, 4 waves
        wmma_gemm_bias<<<grid, 128, 0, stream>>>(fin, w[l], bb[l], fo, NB * NV, ci[l], N);
        conv_aggregate_kernel<<<NB * NV, 256, 0, stream>>>(
            fo, dn, idx, dirsn[l], fot, co[l], (l < 4) ? 1 : 0);
        float* tmp = fin; fin = fot; fot = tmp;
    }
    // fin == fmA == fm4 (2048 x 1024)

    // 5. global max pool -> d_out[0..2048)
    float* fg = (float*)d_out;
    maxpool_kernel<<<8, 256, 0, stream>>>(fin, fg);

    // 6. heads: cls -> d_out[2048..2054), proj(L2-normalized) -> d_out[2054..2310)
    head_kernel<<<NB, 256, 0, stream>>>(fg,
        (const float*)d_in[14], (const float*)d_in[15], (const float*)d_in[16],
        (const float*)d_in[17], (const float*)d_in[18], (const float*)d_in[19],
        (const float*)d_in[20], (const float*)d_in[21],
        fg + 2048, 3, 0);
    head_kernel<<<NB, 256, 0, stream>>>(fg,
        (const float*)d_in[22], (const float*)d_in[23], (const float*)d_in[24],
        (const float*)d_in[25], (const float*)d_in[26], (const float*)d_in[27],
        (const float*)d_in[28], (const float*)d_in[29],
        fg + 2048 + 6, 128, 1);
}